// connectionBlock_85401129713890
// MI455X (gfx1250) — compile-verified
//
#include <hip/hip_runtime.h>
#include <hip/hip_bf16.h>
#include <math.h>

#define Bb 4
#define Tt 2048
#define Cc 1024
#define Hh 4
#define Dd 256
#define Mm (Bb*Tt)

typedef float v2f __attribute__((ext_vector_type(2)));
typedef float v8f __attribute__((ext_vector_type(8)));

// fp32 WMMA: D(16x16) = A(16x4) * B(4x16) + C, exact fp32 numerics.
__device__ __forceinline__ v8f wmma4(v2f a, v2f b, v8f c) {
  return __builtin_amdgcn_wmma_f32_16x16x4_f32(false, a, false, b, (short)0, c,
                                               false, false);
}

__device__ __forceinline__ float block_reduce_sum(float v, float* red) {
  int tid = threadIdx.x;
  red[tid] = v;
  __syncthreads();
  for (int off = 128; off > 0; off >>= 1) {
    if (tid < off) red[tid] += red[tid + off];
    __syncthreads();
  }
  float r = red[0];
  __syncthreads();
  return r;
}

// LayerNorm over last dim (C) of a (rows, C) matrix; w,b flat length C.
__global__ __launch_bounds__(256) void ln_row_kernel(
    const float* __restrict__ in, const float* __restrict__ w,
    const float* __restrict__ b, float* __restrict__ out) {
  __shared__ float red[256];
  size_t row = blockIdx.x;
  const float* ir = in + row * Cc;
  float v[4];
  float s = 0.f;
#pragma unroll
  for (int i = 0; i < 4; ++i) { v[i] = ir[threadIdx.x + 256 * i]; s += v[i]; }
  float mu = block_reduce_sum(s, red) * (1.f / Cc);
  float s2 = 0.f;
#pragma unroll
  for (int i = 0; i < 4; ++i) { float d = v[i] - mu; s2 += d * d; }
  float var = block_reduce_sum(s2, red) * (1.f / Cc);
  float inv = rsqrtf(var + 1e-5f);
#pragma unroll
  for (int i = 0; i < 4; ++i) {
    int c = threadIdx.x + 256 * i;
    out[row * Cc + c] = (v[i] - mu) * inv * w[c] + b[c];
  }
}

// g = dwconv3(h) * mask, then channel-LN.  h layout (B,T,C), conv over T.
__global__ __launch_bounds__(256) void dwconv_ln_kernel(
    const float* __restrict__ h, const float* __restrict__ cw,
    const unsigned char* __restrict__ mask, const float* __restrict__ lw,
    const float* __restrict__ lb, float* __restrict__ out) {
  __shared__ float red[256];
  int bt = blockIdx.x;
  int t = bt & (Tt - 1);
  float mf = mask[bt] ? 1.f : 0.f;
  const float* h0 = h + (size_t)bt * Cc;
  float v[4];
  float s = 0.f;
#pragma unroll
  for (int i = 0; i < 4; ++i) {
    int c = threadIdx.x + 256 * i;
    float acc = h0[c] * cw[c * 3 + 1];
    if (t > 0) acc += h0[c - Cc] * cw[c * 3 + 0];
    if (t < Tt - 1) acc += h0[c + Cc] * cw[c * 3 + 2];
    v[i] = acc * mf;
    s += v[i];
  }
  float mu = block_reduce_sum(s, red) * (1.f / Cc);
  float s2 = 0.f;
#pragma unroll
  for (int i = 0; i < 4; ++i) { float d = v[i] - mu; s2 += d * d; }
  float var = block_reduce_sum(s2, red) * (1.f / Cc);
  float inv = rsqrtf(var + 1e-5f);
#pragma unroll
  for (int i = 0; i < 4; ++i) {
    int c = threadIdx.x + 256 * i;
    out[(size_t)bt * Cc + c] = (v[i] - mu) * inv * lw[c] + lb[c];
  }
}

// y2 = gelu_exact(dwconv3(y1) + bias)
__global__ __launch_bounds__(256) void dwconv_gelu_kernel(
    const float* __restrict__ y1, const float* __restrict__ cw,
    const float* __restrict__ cb, float* __restrict__ out) {
  int bt = blockIdx.x;
  int t = bt & (Tt - 1);
  const float* h0 = y1 + (size_t)bt * Cc;
#pragma unroll
  for (int i = 0; i < 4; ++i) {
    int c = threadIdx.x + 256 * i;
    float acc = h0[c] * cw[c * 3 + 1] + cb[c];
    if (t > 0) acc += h0[c - Cc] * cw[c * 3 + 0];
    if (t < Tt - 1) acc += h0[c + Cc] * cw[c * 3 + 2];
    float g = 0.5f * acc * (1.0f + erff(acc * 0.70710678118654752f));
    out[(size_t)bt * Cc + c] = g;
  }
}

// out(M,N) = A(M,K) @ W(N,K)^T + bias, fp32 WMMA 16x16x4.
// One wave computes a 32x32 output tile (2x2 subtiles of 16x16), reusing each
// A/B fragment across two WMMAs: 4 loads -> 4 WMMAs per K-step.
// MODE 0: plain; MODE 1: (gemm+bias)*mask[row] + resid; MODE 2: gemm+bias+resid
template <int MODE>
__global__ __launch_bounds__(256) void gemm_kernel(
    const float* __restrict__ A, const float* __restrict__ W,
    const float* __restrict__ bias, const float* __restrict__ resid,
    const unsigned char* __restrict__ mask, float* __restrict__ out) {
  const int K = Cc, N = Cc;
  int wave = threadIdx.x >> 5;
  int lane = threadIdx.x & 31;
  int g = blockIdx.x * 8 + wave;   // tile id; (M/32)*(N/32) = 256*32 tiles
  int mt = g >> 5;                 // 0..255
  int nt = g & 31;                 // 0..31
  int gi = lane >> 4;              // lane-half
  int ln = lane & 15;
  const float* arow0 = A + (size_t)(mt * 32 + ln) * K + 2 * gi;
  const float* arow1 = arow0 + (size_t)16 * K;
  const float* wrow0 = W + (size_t)(nt * 32 + ln) * K + 2 * gi;
  const float* wrow1 = wrow0 + (size_t)16 * K;
  v8f acc00 = {0.f,0.f,0.f,0.f,0.f,0.f,0.f,0.f};
  v8f acc01 = acc00, acc10 = acc00, acc11 = acc00;
#pragma unroll 4
  for (int k4 = 0; k4 < K / 4; ++k4) {
    v2f a0 = *(const v2f*)(arow0 + k4 * 4);
    v2f a1 = *(const v2f*)(arow1 + k4 * 4);
    v2f b0 = *(const v2f*)(wrow0 + k4 * 4);
    v2f b1 = *(const v2f*)(wrow1 + k4 * 4);
    acc00 = wmma4(a0, b0, acc00);
    acc01 = wmma4(a0, b1, acc01);
    acc10 = wmma4(a1, b0, acc10);
    acc11 = wmma4(a1, b1, acc11);
  }
  v8f accs[2][2] = {{acc00, acc01}, {acc10, acc11}};
#pragma unroll
  for (int mi = 0; mi < 2; ++mi) {
#pragma unroll
    for (int ni = 0; ni < 2; ++ni) {
      int col = nt * 32 + ni * 16 + ln;
      float bv = bias[col];
#pragma unroll
      for (int r = 0; r < 8; ++r) {
        int row = mt * 32 + mi * 16 + r + 8 * gi;
        float v = accs[mi][ni][r] + bv;
        if (MODE == 1) {
          float mf = mask[row] ? 1.f : 0.f;
          v = v * mf + resid[(size_t)row * N + col];
        } else if (MODE == 2) {
          v += resid[(size_t)row * N + col];
        }
        out[(size_t)row * N + col] = v;
      }
    }
  }
}

// Flash attention: one wave per (b, h, 16-row tile). q,k,v,o in (B,T,C),
// head h occupies channels [h*D, (h+1)*D).  Softmax zeroes masked cols, so
// the v*mask multiply of the reference is implicit.
__global__ __launch_bounds__(32) void attn_kernel(
    const float* __restrict__ q, const float* __restrict__ k,
    const float* __restrict__ v, const unsigned char* __restrict__ mask,
    float* __restrict__ o) {
  __shared__ float p_lds[16 * 16];
  int i0 = blockIdx.x * 16;
  int hh = blockIdx.y;
  int b = blockIdx.z;
  int hD = hh * Dd;
  int lane = threadIdx.x;
  int g = lane >> 4, ln = lane & 15;
  const float* qb = q + (size_t)b * Tt * Cc;
  const float* kb = k + (size_t)b * Tt * Cc;
  const float* vb = v + (size_t)b * Tt * Cc;
  const unsigned char* mb = mask + b * Tt;

  v8f oa[16];
#pragma unroll
  for (int i = 0; i < 16; ++i) oa[i] = (v8f){0.f,0.f,0.f,0.f,0.f,0.f,0.f,0.f};
  float rmax[8], rsum[8];
#pragma unroll
  for (int r = 0; r < 8; ++r) { rmax[r] = -__builtin_inff(); rsum[r] = 0.f; }

  const float* qrow = qb + (size_t)(i0 + ln) * Cc + hD + 2 * g;

  for (int jt = 0; jt < Tt / 16; ++jt) {
    int j0 = jt * 16;
    const float* krow = kb + (size_t)(j0 + ln) * Cc + hD + 2 * g;
    // Two interleaved accumulators: halve the dependent-WMMA chain depth.
    v8f se = {0.f, 0.f, 0.f, 0.f, 0.f, 0.f, 0.f, 0.f};
    v8f so = se;
#pragma unroll 8
    for (int kk = 0; kk < Dd / 4; kk += 2) {
      v2f a0 = *(const v2f*)(qrow + kk * 4);
      v2f b0 = *(const v2f*)(krow + kk * 4);
      v2f a1 = *(const v2f*)(qrow + kk * 4 + 4);
      v2f b1 = *(const v2f*)(krow + kk * 4 + 4);
      se = wmma4(a0, b0, se);
      so = wmma4(a1, b1, so);
    }
    v8f s = se + so;
    bool valid = mb[j0 + ln] != 0;
    float cf[8];
#pragma unroll
    for (int r = 0; r < 8; ++r) {
      float sv = valid ? s[r] * 0.0625f : -__builtin_inff();
      float m = sv;
#pragma unroll
      for (int off = 8; off > 0; off >>= 1)
        m = fmaxf(m, __shfl_xor(m, off, 32));
      float nm = fmaxf(rmax[r], m);
      float c = (rmax[r] == -__builtin_inff()) ? 0.f : __expf(rmax[r] - nm);
      float p = __expf(sv - nm);
      float ps = p;
#pragma unroll
      for (int off = 8; off > 0; off >>= 1) ps += __shfl_xor(ps, off, 32);
      rsum[r] = rsum[r] * c + ps;
      rmax[r] = nm;
      cf[r] = c;
      p_lds[(r + 8 * g) * 16 + ln] = p;
    }
#pragma unroll
    for (int dt = 0; dt < 16; ++dt)
#pragma unroll
      for (int r = 0; r < 8; ++r) oa[dt][r] *= cf[r];
    __syncthreads();
    v2f pa[4];
#pragma unroll
    for (int ks = 0; ks < 4; ++ks)
      pa[ks] = *(const v2f*)&p_lds[ln * 16 + ks * 4 + 2 * g];
    __syncthreads();
#pragma unroll
    for (int dt = 0; dt < 16; ++dt) {
      int colv = hD + dt * 16 + ln;
#pragma unroll
      for (int ks = 0; ks < 4; ++ks) {
        int kr = j0 + ks * 4 + 2 * g;
        v2f bf;
        bf.x = vb[(size_t)kr * Cc + colv];
        bf.y = vb[(size_t)(kr + 1) * Cc + colv];
        oa[dt] = wmma4(pa[ks], bf, oa[dt]);
      }
    }
  }
#pragma unroll
  for (int dt = 0; dt < 16; ++dt) {
    int colv = hD + dt * 16 + ln;
#pragma unroll
    for (int r = 0; r < 8; ++r) {
      int row = i0 + r + 8 * g;
      o[((size_t)b * Tt + row) * Cc + colv] = oa[dt][r] / rsum[r];
    }
  }
}

extern "C" void kernel_launch(void* const* d_in, const int* in_sizes, int n_in,
                              void* d_out, int out_size, void* d_ws,
                              size_t ws_size, hipStream_t stream) {
  const float* x        = (const float*)d_in[0];
  const unsigned char* mask = (const unsigned char*)d_in[1];
  const float* norm1_w  = (const float*)d_in[2];
  const float* norm1_b  = (const float*)d_in[3];
  const float* qconv_w  = (const float*)d_in[4];
  const float* kconv_w  = (const float*)d_in[5];
  const float* vconv_w  = (const float*)d_in[6];
  const float* qnorm_w  = (const float*)d_in[7];
  const float* qnorm_b  = (const float*)d_in[8];
  const float* knorm_w  = (const float*)d_in[9];
  const float* knorm_b  = (const float*)d_in[10];
  const float* vnorm_w  = (const float*)d_in[11];
  const float* vnorm_b  = (const float*)d_in[12];
  const float* q1_w = (const float*)d_in[13];
  const float* q1_b = (const float*)d_in[14];
  const float* k1_w = (const float*)d_in[15];
  const float* k1_b = (const float*)d_in[16];
  const float* v1_w = (const float*)d_in[17];
  const float* v1_b = (const float*)d_in[18];
  const float* proj_w = (const float*)d_in[19];
  const float* proj_b = (const float*)d_in[20];
  const float* norm2_w = (const float*)d_in[21];
  const float* norm2_b = (const float*)d_in[22];
  const float* lin1_w = (const float*)d_in[23];
  const float* lin1_b = (const float*)d_in[24];
  const float* tc_w = (const float*)d_in[25];
  const float* tc_b = (const float*)d_in[26];
  const float* lin2_w = (const float*)d_in[27];
  const float* lin2_b = (const float*)d_in[28];
  float* out = (float*)d_out;

  const size_t S = (size_t)Mm * Cc;   // 8M floats = 32MB per slot
  float* s0 = (float*)d_ws;           // h, then attention output
  float* s1 = s0 + S;                 // act (dwconv+ln out), then h2
  float* s2 = s1 + S;                 // q, then y1
  float* s3 = s2 + S;                 // k, then y2
  float* s4 = s3 + S;                 // v, then x_res

  dim3 blk(256);
  dim3 rows(Mm);
  dim3 gemm_grid(1024);               // (M/32)*(N/32)=8192 tiles / 8 waves per block

  // h = channel LN of x
  ln_row_kernel<<<rows, blk, 0, stream>>>(x, norm1_w, norm1_b, s0);

  // q
  dwconv_ln_kernel<<<rows, blk, 0, stream>>>(s0, qconv_w, mask, qnorm_w, qnorm_b, s1);
  gemm_kernel<0><<<gemm_grid, blk, 0, stream>>>(s1, q1_w, q1_b, (const float*)0,
                                                (const unsigned char*)0, s2);
  // k
  dwconv_ln_kernel<<<rows, blk, 0, stream>>>(s0, kconv_w, mask, knorm_w, knorm_b, s1);
  gemm_kernel<0><<<gemm_grid, blk, 0, stream>>>(s1, k1_w, k1_b, (const float*)0,
                                                (const unsigned char*)0, s3);
  // v
  dwconv_ln_kernel<<<rows, blk, 0, stream>>>(s0, vconv_w, mask, vnorm_w, vnorm_b, s1);
  gemm_kernel<0><<<gemm_grid, blk, 0, stream>>>(s1, v1_w, v1_b, (const float*)0,
                                                (const unsigned char*)0, s4);

  // attention -> s0
  attn_kernel<<<dim3(Tt / 16, Hh, Bb), dim3(32), 0, stream>>>(s2, s3, s4, mask, s0);

  // x_res = x + mask * (proj(attn) + b)  -> s4
  gemm_kernel<1><<<gemm_grid, blk, 0, stream>>>(s0, proj_w, proj_b, x, mask, s4);

  // FFN branch: h2 = LN(x_res); y1 = lin1; y2 = gelu(dwconv(y1)+b); out = x_res + lin2(y2)
  ln_row_kernel<<<rows, blk, 0, stream>>>(s4, norm2_w, norm2_b, s1);
  gemm_kernel<0><<<gemm_grid, blk, 0, stream>>>(s1, lin1_w, lin1_b, (const float*)0,
                                                (const unsigned char*)0, s2);
  dwconv_gelu_kernel<<<rows, blk, 0, stream>>>(s2, tc_w, tc_b, s3);
  gemm_kernel<2><<<gemm_grid, blk, 0, stream>>>(s3, lin2_w, lin2_b, s4,
                                                (const unsigned char*)0, out);

  (void)in_sizes; (void)n_in; (void)out_size; (void)ws_size;
}